// VectorQuantizer_73486890434653
// MI455X (gfx1250) — compile-verified
//
#include <hip/hip_runtime.h>

typedef __attribute__((ext_vector_type(2))) float v2f;
typedef __attribute__((ext_vector_type(8))) float v8f;

#define TT 4096   // T
#define DD 64     // D
#define KK 1024   // codebook size
// tokens = 16*4096 = 65536; 16 tokens/wave -> 4096 wave tiles
// block = 512 threads (16 waves), grid = 256 blocks

// ---------------- workspace layout ----------------
// [0, 262144)       cbG : grouped codebook, v2f units:
//                   [g=n/16][j=0..31][n%16] ; group stride 4096 B, row 128 B
// [262144, 266240)  hes : 1024 floats (0.5*||e||^2)
// [266240, 528384)  idx : 65536 ints
//
// ---------------- LDS layout (dynamic, 266240 B) ----------------
// [0, 262144)       cbG staged verbatim (linear copy)
// [262144, 266240)  hes staged
#define LDS_HES_B 262144
#define LDS_TOTAL 266240

__global__ void __launch_bounds__(256)
vq_prep(const float* __restrict__ cb, float* __restrict__ cbG,
        float* __restrict__ hes) {
  int n = blockIdx.x * blockDim.x + threadIdx.x;
  if (n >= KK) return;
  const float* row = cb + n * DD;
  float s = 0.f;
  v2f* cp = (v2f*)cbG;
  int base = (n >> 4) * 512 + (n & 15);   // v2f units: g*512 + ln
#pragma unroll
  for (int j = 0; j < 32; ++j) {
    float x = row[2 * j];
    float y = row[2 * j + 1];
    s += x * x + y * y;
    v2f p;
    p.x = x;
    p.y = y;
    cp[base + j * 16] = p;                // [g][j][ln]
  }
  hes[n] = 0.5f * s;
}

__global__ void __launch_bounds__(512)
vq_encode(const float* __restrict__ z, const float* __restrict__ cbG,
          const float* __restrict__ hes, int* __restrict__ idx) {
  extern __shared__ char smem[];

  // ---- stage grouped codebook + hes into LDS (linear, coalesced) ----
  {
    int tid = threadIdx.x;                    // 0..511
    const float4* src4 = (const float4*)cbG;  // 16384 float4
    float4* dst4 = (float4*)smem;
#pragma unroll
    for (int it = 0; it < 32; ++it) {
      dst4[it * 512 + tid] = src4[it * 512 + tid];
    }
    float* shes = (float*)(smem + LDS_HES_B);
    for (int i = tid; i < KK; i += 512) shes[i] = hes[i];
  }
  __syncthreads();

  const int lane = threadIdx.x & 31;
  const int wave = threadIdx.x >> 5;            // 0..15
  const int tile = blockIdx.x * 16 + wave;      // 0..4095
  const int token0 = tile << 4;                 // first of 16 tokens
  const int b = token0 >> 12;                   // batch (T=4096)
  const int t0 = token0 & (TT - 1);
  const int hi = lane >> 4;                     // half-wave select
  const int ln = lane & 15;

  // ---- preload A (16 tokens x 64 d), ISA 32-bit A 16x4 layout per chunk:
  // lanes 0-15: K=4k+0,4k+1 ; lanes 16-31: K=4k+2,4k+3 ; M = ln
  const float* zb = z + (size_t)b * DD * TT + t0 + ln;
  float a[32];
#pragma unroll
  for (int kk = 0; kk < 16; ++kk) {
    int d0 = 4 * kk + 2 * hi;
    a[2 * kk]     = zb[(size_t)d0 * TT];
    a[2 * kk + 1] = zb[(size_t)(d0 + 1) * TT];
  }
  // Pin A in VGPRs now: resolves all global-load waits BEFORE the main loop
  // (otherwise the compiler re-emits a ladder of s_wait_loadcnt every iter).
#pragma unroll
  for (int i = 0; i < 32; ++i) {
    asm volatile("" : "+v"(a[i]));
  }

  float best[8];
  int bestn[8];
#pragma unroll
  for (int e = 0; e < 8; ++e) { best[e] = -3.4e38f; bestn[e] = 0; }

  // Per-lane LDS bases: group g advances by 2 per iteration (8192 B);
  // all per-kk row offsets are immediates (kk*256, max 3840).
  const char* sb0 = smem + hi * 128 + ln * 8;   // tile 0 (group 2*nt)
  const char* sb1 = sb0 + 4096;                 // tile 1 (group 2*nt+1)
  const float* shp = (const float*)(smem + LDS_HES_B) + ln;  // +32 per iter

  for (int nt = 0; nt < KK / 32; ++nt) {
    const int n0 = nt * 32 + ln;                // tile 0 column
    const int n1 = n0 + 16;                     // tile 1 column
    const float h0 = shp[0];                    // needed only at tile end
    const float h1 = shp[16];
    shp += 32;

    // Buffer ALL B pairs first (16 ds_load_2addr_b64, clause-able)...
    v2f bb0[16], bb1[16];
#pragma unroll
    for (int kk = 0; kk < 16; ++kk) {
      bb0[kk] = *(const v2f*)(sb0 + kk * 256);  // conflict-free banks
      bb1[kk] = *(const v2f*)(sb1 + kk * 256);
    }
    // ...and PIN them in distinct VGPRs: forces a single s_wait_dscnt here
    // (one exposed LDS latency/iter) instead of the scheduler recycling one
    // quad with a dscnt==0 stall before every other WMMA.
#pragma unroll
    for (int kk = 0; kk < 16; ++kk) {
      asm volatile("" : "+v"(bb0[kk]), "+v"(bb1[kk]));
    }
    sb0 += 8192;
    sb1 += 8192;

    v8f c0 = {};                                // inline-0 SRC2 on first WMMA
    v8f c1 = {};
#pragma unroll
    for (int kk = 0; kk < 16; ++kk) {
      v2f av;
      av.x = a[2 * kk];
      av.y = a[2 * kk + 1];
      c0 = __builtin_amdgcn_wmma_f32_16x16x4_f32(
          false, av, false, bb0[kk], (short)0, c0, false, false);
      c1 = __builtin_amdgcn_wmma_f32_16x16x4_f32(
          false, av, false, bb1[kk], (short)0, c1, false, false);
    }

    // bias applied here (score = z.e - 0.5||e||^2); candidate n increasing,
    // '>' keeps the smallest index on exact ties (matches argmin semantics)
#pragma unroll
    for (int e = 0; e < 8; ++e) {
      float s0 = c0[e] - h0;
      float s1 = c1[e] - h1;
      if (s0 > best[e]) { best[e] = s0; bestn[e] = n0; }
      if (s1 > best[e]) { best[e] = s1; bestn[e] = n1; }
    }
  }

  // ---- cross-lane argmax within each 16-lane half (rows 0-7 in lanes 0-15,
  // rows 8-15 in lanes 16-31, per C/D VGPR layout). Tie -> smaller index.
#pragma unroll
  for (int mask = 1; mask <= 8; mask <<= 1) {
#pragma unroll
    for (int e = 0; e < 8; ++e) {
      float ov = __shfl_xor(best[e], mask, 32);
      int oi = __shfl_xor(bestn[e], mask, 32);
      if (ov > best[e] || (ov == best[e] && oi < bestn[e])) {
        best[e] = ov;
        bestn[e] = oi;
      }
    }
  }

  // lanes 0 and 16 hold reduced results for rows 0-7 / 8-15
  if (ln == 0) {
    int base = token0 + hi * 8;
    int4 b0 = make_int4(bestn[0], bestn[1], bestn[2], bestn[3]);
    int4 b1 = make_int4(bestn[4], bestn[5], bestn[6], bestn[7]);
    *(int4*)(idx + base) = b0;
    *(int4*)(idx + base + 4) = b1;
  }
}

__global__ void __launch_bounds__(256)
vq_gather(const float* __restrict__ cb, const int* __restrict__ idx,
          float* __restrict__ out) {
  int gid = blockIdx.x * blockDim.x + threadIdx.x;   // = b*D*T + d*T + t
  int t = gid & (TT - 1);
  int d = (gid >> 12) & (DD - 1);
  int b = gid >> 18;
  int id = idx[b * TT + t];                          // coalesced
  out[gid] = cb[id * DD + d];                        // gather (cache-resident)
}

extern "C" void kernel_launch(void* const* d_in, const int* in_sizes, int n_in,
                              void* d_out, int out_size, void* d_ws,
                              size_t ws_size, hipStream_t stream) {
  const float* z = (const float*)d_in[0];    // [16, 64, 4096]
  const float* cb = (const float*)d_in[1];   // [1024, 64]
  char* ws = (char*)d_ws;
  float* cbG = (float*)ws;
  float* hes = (float*)(ws + 262144);
  int* idx = (int*)(ws + 266240);

  vq_prep<<<4, 256, 0, stream>>>(cb, cbG, hes);
  vq_encode<<<256, 512, LDS_TOTAL, stream>>>(z, cbG, hes, idx);
  vq_gather<<<16384, 256, 0, stream>>>(cb, idx, (float*)d_out);
}